// DHPF_80771154968958
// MI455X (gfx1250) — compile-verified
//
#include <hip/hip_runtime.h>
#include <hip/hip_bf16.h>

// DFT-as-matmul high-pass filter for MI455X (gfx1250).
// FFT2 computed as two complex 256x256 matmuls per image using
// V_WMMA_F32_16X16X4_F32 (full fp32 matrix pipe, wave32, 16x16 tiles).

typedef float v2f __attribute__((ext_vector_type(2)));
typedef float v8f __attribute__((ext_vector_type(8)));

#define HW 256
#define IMG_ELEMS (HW * HW)          // 65536
#define NIMG 512                     // 8 batches * 64 channels
#define TILES_PER_IMG 256            // (256/16)^2
#define WAVES_PER_BLOCK 8
#define PASS_BLOCKS ((NIMG * TILES_PER_IMG) / WAVES_PER_BLOCK)  // 16384
#define PLANE_ELEMS ((size_t)NIMG * IMG_ELEMS)                  // 33,554,432 floats

static __device__ __forceinline__ v8f wmma_f32(v2f a, v2f b, v8f c) {
  // D = A(16x4) * B(4x16) + C, fp32 throughout.
  return __builtin_amdgcn_wmma_f32_16x16x4_f32(
      /*neg_a=*/false, a, /*neg_b=*/false, b,
      /*c_mod=*/(short)0, c, /*reuse_a=*/false, /*reuse_b=*/false);
}

// Minimal box cutoff containing unshifted frequency (r, c); matches the
// reference's _min_cutoff_map evaluated at fftshifted coordinates.
static __device__ __forceinline__ int minc_of(int r, int c) {
  int dr = ((r + 128) & 255) - 128;
  int dc = ((c + 128) & 255) - 128;
  int mr = max(-dr, dr + 1);
  int mc = max(-dc, dc + 1);
  return max(mr, mc);
}

// ---------------------------------------------------------------- twiddles
// Ct[n*256+k] = cos(2*pi*n*k/256), St[n*256+k] = -sin(2*pi*n*k/256)
// (forward DFT matrix W = Ct + i*St; symmetric). Angle reduced mod 256 exactly.
__global__ void __launch_bounds__(256) dhpf_init_twiddle(float* __restrict__ Ct,
                                                         float* __restrict__ St) {
  const int n = blockIdx.x;
  const int k = threadIdx.x;
  const int p = (n * k) & 255;
  const float ang = -6.28318530717958647692f * (float)p * (1.0f / 256.0f);
  float s, c;
  __sincosf(ang, &s, &c);
  Ct[n * HW + k] = c;
  St[n * HW + k] = s;   // sin(-x) = -sin(x)
}

// ---------------------------------------------------------------- pass 1
// Row DFT: Y = X * W  (X real).  Yre = X*Ct, Yim = X*St.
__global__ void __launch_bounds__(256) dhpf_pass1(const float* __restrict__ x,
                                                  const float* __restrict__ Ct,
                                                  const float* __restrict__ St,
                                                  float* __restrict__ Yre,
                                                  float* __restrict__ Yim) {
  const int wave = blockIdx.x * (blockDim.x >> 5) + (threadIdx.x >> 5);
  const int img  = wave >> 8;
  const int t    = wave & 255;
  const int tm   = (t >> 4) << 4;
  const int tn   = (t & 15) << 4;
  const int lane = threadIdx.x & 31;
  const int la   = lane & 15;
  const int kb   = (lane >> 4) << 1;

  const float* X = x + (size_t)img * IMG_ELEMS;
  v8f accR = {0.f, 0.f, 0.f, 0.f, 0.f, 0.f, 0.f, 0.f};
  v8f accI = accR;
#pragma unroll 2
  for (int k = 0; k < HW; k += 4) {
    const int ka = k + kb;
    v2f a = *(const v2f*)(X + (size_t)(tm + la) * HW + ka);   // A: X rows, contiguous
    v2f bc, bs;                                               // B: W columns, strided
    bc.x = Ct[ka * HW + tn + la];
    bc.y = Ct[(ka + 1) * HW + tn + la];
    bs.x = St[ka * HW + tn + la];
    bs.y = St[(ka + 1) * HW + tn + la];
    accR = wmma_f32(a, bc, accR);
    accI = wmma_f32(a, bs, accI);
  }
  const int moff = (lane >> 4) << 3;
  float* yr = Yre + (size_t)img * IMG_ELEMS;
  float* yi = Yim + (size_t)img * IMG_ELEMS;
#pragma unroll
  for (int v = 0; v < 8; ++v) {
    const size_t o = (size_t)(tm + moff + v) * HW + tn + la;
    yr[o] = accR[v];
    yi[o] = accI[v];
  }
}

// ---------------------------------------------------------------- pass 2
// Column DFT: Z = W * Y.  Zre = Ct*Yr - St*Yi, Zim = Ct*Yi + St*Yr.
__global__ void __launch_bounds__(256) dhpf_pass2(const float* __restrict__ Yre,
                                                  const float* __restrict__ Yim,
                                                  const float* __restrict__ Ct,
                                                  const float* __restrict__ St,
                                                  float* __restrict__ Zre,
                                                  float* __restrict__ Zim) {
  const int wave = blockIdx.x * (blockDim.x >> 5) + (threadIdx.x >> 5);
  const int img  = wave >> 8;
  const int t    = wave & 255;
  const int tm   = (t >> 4) << 4;
  const int tn   = (t & 15) << 4;
  const int lane = threadIdx.x & 31;
  const int la   = lane & 15;
  const int kb   = (lane >> 4) << 1;

  const float* yr = Yre + (size_t)img * IMG_ELEMS;
  const float* yi = Yim + (size_t)img * IMG_ELEMS;
  v8f p1 = {0.f, 0.f, 0.f, 0.f, 0.f, 0.f, 0.f, 0.f};
  v8f p2 = p1, p3 = p1, p4 = p1;
#pragma unroll 2
  for (int k = 0; k < HW; k += 4) {
    const int ka = k + kb;
    v2f ac = *(const v2f*)(Ct + (size_t)(tm + la) * HW + ka);  // A: W rows (symmetric)
    v2f as = *(const v2f*)(St + (size_t)(tm + la) * HW + ka);
    v2f br, bi;
    br.x = yr[ka * HW + tn + la];
    br.y = yr[(ka + 1) * HW + tn + la];
    bi.x = yi[ka * HW + tn + la];
    bi.y = yi[(ka + 1) * HW + tn + la];
    p1 = wmma_f32(ac, br, p1);
    p2 = wmma_f32(as, bi, p2);
    p3 = wmma_f32(ac, bi, p3);
    p4 = wmma_f32(as, br, p4);
  }
  const int moff = (lane >> 4) << 3;
  float* zr = Zre + (size_t)img * IMG_ELEMS;
  float* zi = Zim + (size_t)img * IMG_ELEMS;
#pragma unroll
  for (int v = 0; v < 8; ++v) {
    const size_t o = (size_t)(tm + moff + v) * HW + tn + la;
    zr[o] = p1[v] - p2[v];
    zi[o] = p3[v] + p4[v];
  }
}

// ---------------------------------------------------------------- cutoff
// Per batch: LDS histogram of |Z|^2 over the 129 box bins (channel 0),
// serial scan for the 40%-energy cutoff.
__global__ void __launch_bounds__(256) dhpf_cutoff(const float* __restrict__ Zre,
                                                   const float* __restrict__ Zim,
                                                   int* __restrict__ cutoffs) {
  __shared__ float bins[129];
  const int b = blockIdx.x;
  const int tid = threadIdx.x;
  if (tid < 129) bins[tid] = 0.0f;
  __syncthreads();
  const float* zr = Zre + (size_t)(b * 64) * IMG_ELEMS;  // channel 0 of batch b
  const float* zi = Zim + (size_t)(b * 64) * IMG_ELEMS;
  for (int i = tid; i < IMG_ELEMS; i += 256) {
    const float re = zr[i], im = zi[i];
    const float m2 = re * re + im * im;
    atomicAdd(&bins[minc_of(i >> 8, i & 255)], m2);
  }
  __syncthreads();
  if (tid == 0) {
    float total = 0.0f;
    for (int j = 0; j < 129; ++j) total += bins[j];
    const float target = 0.4f * total;
    float cum = 0.0f;
    int cut = 5;
    bool found = false;
    for (int j = 0; j < 128; ++j) {         // reference scans cum[1:128)
      cum += bins[j];
      if (j >= 1 && !found && cum >= target) { cut = j; found = true; }
    }
    cutoffs[b] = cut;
  }
}

// ---------------------------------------------------------------- pass 3
// Masked inverse column DFT: U = conj(W) * (Z .* keep).
// Ure = Ct*Zr' + St*Zi', Uim = Ct*Zi' - St*Zr'.  (1/N folded into pass 4)
__global__ void __launch_bounds__(256) dhpf_pass3(const float* __restrict__ Zre,
                                                  const float* __restrict__ Zim,
                                                  const float* __restrict__ Ct,
                                                  const float* __restrict__ St,
                                                  const int* __restrict__ cutoffs,
                                                  float* __restrict__ Ure,
                                                  float* __restrict__ Uim) {
  const int wave = blockIdx.x * (blockDim.x >> 5) + (threadIdx.x >> 5);
  const int img  = wave >> 8;
  const int t    = wave & 255;
  const int tm   = (t >> 4) << 4;
  const int tn   = (t & 15) << 4;
  const int lane = threadIdx.x & 31;
  const int la   = lane & 15;
  const int kb   = (lane >> 4) << 1;
  const int cut  = cutoffs[img >> 6];

  const float* zr = Zre + (size_t)img * IMG_ELEMS;
  const float* zi = Zim + (size_t)img * IMG_ELEMS;
  v8f p1 = {0.f, 0.f, 0.f, 0.f, 0.f, 0.f, 0.f, 0.f};
  v8f p2 = p1, p3 = p1, p4 = p1;
  const int nglob = tn + la;
#pragma unroll 2
  for (int k = 0; k < HW; k += 4) {
    const int ka = k + kb;
    v2f ac = *(const v2f*)(Ct + (size_t)(tm + la) * HW + ka);
    v2f as = *(const v2f*)(St + (size_t)(tm + la) * HW + ka);
    const float m0 = (minc_of(ka, nglob) > cut) ? 1.0f : 0.0f;
    const float m1 = (minc_of(ka + 1, nglob) > cut) ? 1.0f : 0.0f;
    v2f br, bi;
    br.x = zr[ka * HW + nglob] * m0;
    br.y = zr[(ka + 1) * HW + nglob] * m1;
    bi.x = zi[ka * HW + nglob] * m0;
    bi.y = zi[(ka + 1) * HW + nglob] * m1;
    p1 = wmma_f32(ac, br, p1);
    p2 = wmma_f32(as, bi, p2);
    p3 = wmma_f32(ac, bi, p3);
    p4 = wmma_f32(as, br, p4);
  }
  const int moff = (lane >> 4) << 3;
  float* ur = Ure + (size_t)img * IMG_ELEMS;
  float* ui = Uim + (size_t)img * IMG_ELEMS;
#pragma unroll
  for (int v = 0; v < 8; ++v) {
    const size_t o = (size_t)(tm + moff + v) * HW + tn + la;
    ur[o] = p1[v] + p2[v];
    ui[o] = p3[v] - p4[v];
  }
}

// ---------------------------------------------------------------- pass 4
// Inverse row DFT + magnitude: Out = |U * conj(W)| / N^2.
// Ore = Ur*Ct + Ui*St, Oim = Ui*Ct - Ur*St.
__global__ void __launch_bounds__(256) dhpf_pass4(const float* __restrict__ Ure,
                                                  const float* __restrict__ Uim,
                                                  const float* __restrict__ Ct,
                                                  const float* __restrict__ St,
                                                  float* __restrict__ out) {
  const int wave = blockIdx.x * (blockDim.x >> 5) + (threadIdx.x >> 5);
  const int img  = wave >> 8;
  const int t    = wave & 255;
  const int tm   = (t >> 4) << 4;
  const int tn   = (t & 15) << 4;
  const int lane = threadIdx.x & 31;
  const int la   = lane & 15;
  const int kb   = (lane >> 4) << 1;

  const float* ur = Ure + (size_t)img * IMG_ELEMS;
  const float* ui = Uim + (size_t)img * IMG_ELEMS;
  v8f p1 = {0.f, 0.f, 0.f, 0.f, 0.f, 0.f, 0.f, 0.f};
  v8f p2 = p1, p3 = p1, p4 = p1;
#pragma unroll 2
  for (int k = 0; k < HW; k += 4) {
    const int ka = k + kb;
    v2f ar = *(const v2f*)(ur + (size_t)(tm + la) * HW + ka);  // A: U rows, contiguous
    v2f ai = *(const v2f*)(ui + (size_t)(tm + la) * HW + ka);
    v2f bc, bs;
    bc.x = Ct[ka * HW + tn + la];
    bc.y = Ct[(ka + 1) * HW + tn + la];
    bs.x = St[ka * HW + tn + la];
    bs.y = St[(ka + 1) * HW + tn + la];
    p1 = wmma_f32(ar, bc, p1);
    p2 = wmma_f32(ai, bs, p2);
    p3 = wmma_f32(ai, bc, p3);
    p4 = wmma_f32(ar, bs, p4);
  }
  const int moff = (lane >> 4) << 3;
  float* o = out + (size_t)img * IMG_ELEMS;
  const float scale = 1.0f / 65536.0f;  // 1/N per inverse stage, N=256
#pragma unroll
  for (int v = 0; v < 8; ++v) {
    const float re = p1[v] + p2[v];
    const float im = p3[v] - p4[v];
    o[(size_t)(tm + moff + v) * HW + tn + la] = sqrtf(re * re + im * im) * scale;
  }
}

// ---------------------------------------------------------------- launch
extern "C" void kernel_launch(void* const* d_in, const int* in_sizes, int n_in,
                              void* d_out, int out_size, void* d_ws, size_t ws_size,
                              hipStream_t stream) {
  (void)in_sizes; (void)n_in; (void)out_size; (void)ws_size;
  const float* x = (const float*)d_in[0];
  float* out = (float*)d_out;

  // Workspace layout:
  //   [0, 256KB)        Ct   (cos table, 256x256 f32)
  //   [256KB, 512KB)    St   (-sin table)
  //   [512KB, 512KB+32) cutoffs (8 ints)
  //   [1MB, +128MB)x4   Yre, Yim, Zre, Zim complex planes (Y reused for U)
  float* Ct = (float*)d_ws;
  float* St = Ct + IMG_ELEMS;
  int* cutoffs = (int*)(St + IMG_ELEMS);
  float* base = (float*)((char*)d_ws + (1u << 20));
  float* Yre = base;
  float* Yim = Yre + PLANE_ELEMS;
  float* Zre = Yim + PLANE_ELEMS;
  float* Zim = Zre + PLANE_ELEMS;

  dhpf_init_twiddle<<<HW, HW, 0, stream>>>(Ct, St);
  dhpf_pass1<<<PASS_BLOCKS, 256, 0, stream>>>(x, Ct, St, Yre, Yim);
  dhpf_pass2<<<PASS_BLOCKS, 256, 0, stream>>>(Yre, Yim, Ct, St, Zre, Zim);
  dhpf_cutoff<<<8, 256, 0, stream>>>(Zre, Zim, cutoffs);
  dhpf_pass3<<<PASS_BLOCKS, 256, 0, stream>>>(Zre, Zim, Ct, St, cutoffs, Yre, Yim);
  dhpf_pass4<<<PASS_BLOCKS, 256, 0, stream>>>(Yre, Yim, Ct, St, out);
}